// GNN_66434554135166
// MI455X (gfx1250) — compile-verified
//
#include <hip/hip_runtime.h>
#include <hip/hip_bf16.h>

// ---------------------------------------------------------------------------
// GAT (3x GATConv + pool + MLP) for MI455X / gfx1250.
// Dense math uses V_WMMA_F32_16X16X4_F32 (exact fp32, wave32). Each wave
// computes an (MT*16)x16 C panel, reusing the B fragment across MT WMMAs per
// K-step. MT is a template parameter chosen to divide tiles_m exactly
// (MT=5 for N=50000 nodes: 3125=5^5 row tiles; MT=4 for the 2048-row MLP),
// so the inner loop has ZERO branches, full EXEC, and no remainder handling:
// per K-step MT x global_load_b64 (A) + 2 x b32 (B) + MT x v_wmma, no spills.
// K is always a multiple of 4 (layer-1 x and W1 padded 30 -> 32).
// Edge softmax/aggregation is scatter/gather bound; node buffers (25.6MB)
// stay L2-resident (192MB L2).
// ---------------------------------------------------------------------------

typedef float v2f __attribute__((ext_vector_type(2)));
typedef float v8f __attribute__((ext_vector_type(8)));

#define NODES   50000
#define NEDGE   800000
#define NGRAPH  2048
#define NODEF   30
#define NODEFP  32        // padded K for layer 1
#define FEAT    128       // HEADS * HID
#define HEADS   4
#define HIDC    32
#define MT_NODE 5         // 3125 row tiles = 5^5  -> divisible by 5
#define MT_MLP  4         // 128 row tiles         -> divisible by 4

// ---- ordered-uint encoding for exact float atomic max ----------------------
__device__ __forceinline__ unsigned fenc(float f) {
  unsigned u = __float_as_uint(f);
  return (u & 0x80000000u) ? ~u : (u | 0x80000000u);
}
__device__ __forceinline__ float fdec(unsigned u) {
  return __uint_as_float((u & 0x80000000u) ? (u & 0x7FFFFFFFu) : ~u);
}
#define ENC_NEG_INF 0x007FFFFFu   // fenc(-inf)

// ---- WMMA fp32 GEMM: C[M,N] = A[M,K] @ W[K,N] (+bias, +relu) ---------------
// Requirements: (M/16) % MT == 0, N % 16 == 0, K % 4 == 0.
// A layout per lane: lanes0-15 hold (row=r0+l, k=k0+{0,1}); lanes16-31 k0+{2,3}.
// B layout: lanes0-15 hold (k=k0+{0,1}, col=c0+l); lanes16-31 k0+{2,3}.
// C/D: vgpr r -> row = r0 + r + 8*(lane>>4), col = c0 + (lane&15).
template <int MT>
__global__ void gat_wmma_gemm_f32(const float* __restrict__ A,
                                  const float* __restrict__ W,
                                  float* __restrict__ C,
                                  const float* __restrict__ bias,
                                  int M, int K, int N, int relu) {
  int wave     = threadIdx.x >> 5;
  int lane     = threadIdx.x & 31;
  int tiles_n  = N >> 4;
  int groups_m = (M >> 4) / MT;             // exact by construction
  int gid = blockIdx.x * (blockDim.x >> 5) + wave;
  if (gid >= groups_m * tiles_n) return;    // whole-wave exit: EXEC all-1 for WMMA
  int rt0 = (gid / tiles_n) * MT;
  int c0  = (gid % tiles_n) << 4;
  int l15 = lane & 15;
  int kb  = (lane >> 4) << 1;               // 0 or 2

  const float* Arow = A + ((size_t)(rt0 << 4) + l15) * K + kb;
  const float* Bp   = W + (size_t)kb * N + c0 + l15;

  v8f acc[MT];
#pragma unroll
  for (int i = 0; i < MT; ++i) acc[i] = (v8f){};

  for (int k0 = 0; k0 < K; k0 += 4) {
    v2f b;
    b.x = Bp[0];
    b.y = Bp[(size_t)N];
    Bp += (size_t)4 * N;
    v2f a[MT];
#pragma unroll
    for (int i = 0; i < MT; ++i)
      a[i] = *(const v2f*)(Arow + (size_t)i * 16 * K + k0);   // global_load_b64
#pragma unroll
    for (int i = 0; i < MT; ++i)
      acc[i] = __builtin_amdgcn_wmma_f32_16x16x4_f32(
          false, a[i], false, b, (short)0, acc[i], false, false);
  }

  int col = c0 + l15;
  float bv = bias ? bias[col] : 0.0f;
#pragma unroll
  for (int i = 0; i < MT; ++i) {
    int rbase = ((rt0 + i) << 4) + ((lane >> 4) << 3);
#pragma unroll
    for (int r = 0; r < 8; ++r) {
      float v = acc[i][r] + bv;
      if (relu) v = v > 0.0f ? v : 0.0f;
      C[(size_t)(rbase + r) * N + col] = v;
    }
  }
}

// ---- pad node features [N,30] -> [N,32] ------------------------------------
__global__ void gat_pad_x(const float* __restrict__ x, float* __restrict__ xp, int N) {
  int t = blockIdx.x * blockDim.x + threadIdx.x;
  if (t >= N * NODEFP) return;
  int n = t >> 5, c = t & 31;
  xp[t] = (c < NODEF) ? x[(size_t)n * NODEF + c] : 0.0f;
}

// ---- pad W1 [30,128] -> [32,128] (zero rows 30..31) ------------------------
__global__ void gat_pad_w1(const float* __restrict__ W, float* __restrict__ Wp) {
  int t = blockIdx.x * blockDim.x + threadIdx.x;
  if (t >= NODEFP * FEAT) return;
  int row = t >> 7;
  Wp[t] = (row < NODEF) ? W[t] : 0.0f;
}

// ---- mean of edge features (8 columns) -------------------------------------
__global__ void gat_mean_edge(const float* __restrict__ ef, float* __restrict__ meanE, int E) {
  __shared__ float sh[256];
  int f = blockIdx.x;                    // 0..7
  float s = 0.0f;
  for (int e = threadIdx.x; e < E; e += blockDim.x) s += ef[(size_t)e * 8 + f];
  sh[threadIdx.x] = s;
  __syncthreads();
  for (int o = 128; o > 0; o >>= 1) {
    if ((int)threadIdx.x < o) sh[threadIdx.x] += sh[threadIdx.x + o];
    __syncthreads();
  }
  if (threadIdx.x == 0) meanE[f] = sh[0] / (float)E;
}

// ---- per-layer prep: Wea[f,h] = sum_c We[f,h*32+c]*ae[h,c]; selfv[h] -------
__global__ void gat_layer_prep(const float* __restrict__ We, const float* __restrict__ ae,
                               const float* __restrict__ meanE,
                               float* __restrict__ Wea, float* __restrict__ selfv) {
  int t = threadIdx.x;                   // 32 threads
  if (t < 32) {
    int f = t >> 2, hh = t & 3;
    float s = 0.0f;
    for (int c = 0; c < HIDC; ++c) s += We[f * FEAT + hh * HIDC + c] * ae[hh * HIDC + c];
    Wea[f * 4 + hh] = s;
  }
  __syncthreads();
  if (t < 4) {
    float s = 0.0f;
    for (int f = 0; f < 8; ++f) s += meanE[f] * Wea[f * 4 + t];
    selfv[t] = s;
  }
}

// ---- node attention dots: al_s/al_d [N,4] ----------------------------------
__global__ void gat_node_dots(const float* __restrict__ h, const float* __restrict__ asrc,
                              const float* __restrict__ adst,
                              float* __restrict__ als, float* __restrict__ ald, int N) {
  int n = blockIdx.x * blockDim.x + threadIdx.x;
  if (n >= N) return;
  const float* hp = h + (size_t)n * FEAT;
#pragma unroll
  for (int hh = 0; hh < HEADS; ++hh) {
    float s = 0.0f, d = 0.0f;
    for (int c = 0; c < HIDC; ++c) {
      float v = hp[hh * HIDC + c];
      s += v * asrc[hh * HIDC + c];
      d += v * adst[hh * HIDC + c];
    }
    als[(size_t)n * 4 + hh] = s;
    ald[(size_t)n * 4 + hh] = d;
  }
}

// ---- al_e = eattr @ Wea ----------------------------------------------------
__global__ void gat_edge_ale(const float* __restrict__ ef, const float* __restrict__ Wea,
                             float* __restrict__ ale, int E) {
  int e = blockIdx.x * blockDim.x + threadIdx.x;
  if (e >= E) return;
  float a0 = 0, a1 = 0, a2 = 0, a3 = 0;
#pragma unroll
  for (int f = 0; f < 8; ++f) {
    float v = ef[(size_t)e * 8 + f];
    a0 += v * Wea[f * 4 + 0];
    a1 += v * Wea[f * 4 + 1];
    a2 += v * Wea[f * 4 + 2];
    a3 += v * Wea[f * 4 + 3];
  }
  ale[(size_t)e * 4 + 0] = a0;
  ale[(size_t)e * 4 + 1] = a1;
  ale[(size_t)e * 4 + 2] = a2;
  ale[(size_t)e * 4 + 3] = a3;
}

// ---- per-layer init --------------------------------------------------------
__global__ void gat_init_layer(float* __restrict__ agg, unsigned* __restrict__ mx,
                               float* __restrict__ den, int N) {
  int t = blockIdx.x * blockDim.x + threadIdx.x;
  if (t < N * FEAT) agg[t] = 0.0f;
  if (t < N * HEADS) { mx[t] = ENC_NEG_INF; den[t] = 0.0f; }
}

// ---- pass 1: raw alpha + leakyrelu + segment max ---------------------------
__global__ void gat_edge_max(const int* __restrict__ ei, const float* __restrict__ als,
                             const float* __restrict__ ald, const float* __restrict__ ale,
                             const float* __restrict__ selfv,
                             float* __restrict__ araw, unsigned* __restrict__ mx,
                             int E, int E2) {
  int e = blockIdx.x * blockDim.x + threadIdx.x;
  if (e >= E2) return;
  int s, d;
  float a[HEADS];
  if (e < E) {
    s = ei[e]; d = ei[E + e];
#pragma unroll
    for (int hh = 0; hh < HEADS; ++hh) a[hh] = ale[(size_t)e * 4 + hh];
  } else {
    s = d = e - E;
#pragma unroll
    for (int hh = 0; hh < HEADS; ++hh) a[hh] = selfv[hh];
  }
#pragma unroll
  for (int hh = 0; hh < HEADS; ++hh) {
    float v = als[(size_t)s * 4 + hh] + ald[(size_t)d * 4 + hh] + a[hh];
    v = v > 0.0f ? v : 0.2f * v;
    araw[(size_t)e * 4 + hh] = v;
    atomicMax(&mx[(size_t)d * 4 + hh], fenc(v));
  }
}

// ---- pass 2: exp(alpha - max) + denominator --------------------------------
__global__ void gat_edge_exp(const int* __restrict__ ei, float* __restrict__ ex,
                             const unsigned* __restrict__ mx, float* __restrict__ den,
                             int E, int E2) {
  int e = blockIdx.x * blockDim.x + threadIdx.x;
  if (e >= E2) return;
  int d = (e < E) ? ei[E + e] : (e - E);
#pragma unroll
  for (int hh = 0; hh < HEADS; ++hh) {
    float v = ex[(size_t)e * 4 + hh];
    float w = __expf(v - fdec(mx[(size_t)d * 4 + hh]));
    ex[(size_t)e * 4 + hh] = w;
    atomicAdd(&den[(size_t)d * 4 + hh], w);
  }
}

// ---- pass 3: out[dst] += h[src] * alpha ------------------------------------
__global__ void gat_edge_msg(const int* __restrict__ ei, const float* __restrict__ h,
                             const float* __restrict__ ex, const float* __restrict__ den,
                             float* __restrict__ out, int E, int E2) {
  long long t = (long long)blockIdx.x * blockDim.x + threadIdx.x;
  if (t >= (long long)E2 * FEAT) return;
  int e = (int)(t >> 7);
  int c = (int)(t & (FEAT - 1));
  int s, d;
  if (e < E) { s = ei[e]; d = ei[E + e]; } else { s = d = e - E; }
  int hh = c >> 5;
  float w = ex[(size_t)e * 4 + hh] / (den[(size_t)d * 4 + hh] + 1e-16f);
  atomicAdd(&out[(size_t)d * FEAT + c], h[(size_t)s * FEAT + c] * w);
}

// ---- bias + relu in place ---------------------------------------------------
__global__ void gat_bias_relu(float* __restrict__ out, const float* __restrict__ b, int N) {
  int t = blockIdx.x * blockDim.x + threadIdx.x;
  if (t >= N * FEAT) return;
  float v = out[t] + b[t & (FEAT - 1)];
  out[t] = v > 0.0f ? v : 0.0f;
}

// ---- pooling ----------------------------------------------------------------
__global__ void gat_zero(float* __restrict__ p, int n) {
  int t = blockIdx.x * blockDim.x + threadIdx.x;
  if (t < n) p[t] = 0.0f;
}
__global__ void gat_pool(const float* __restrict__ h, const int* __restrict__ batch,
                         float* __restrict__ g, int N) {
  int t = blockIdx.x * blockDim.x + threadIdx.x;
  if (t >= N * FEAT) return;
  int n = t >> 7, c = t & (FEAT - 1);
  atomicAdd(&g[(size_t)batch[n] * FEAT + c], h[t]);
}

// ---- final 32 -> 1 layer ----------------------------------------------------
__global__ void gat_mlp_out(const float* __restrict__ g2, const float* __restrict__ lw3,
                            const float* __restrict__ lb3, float* __restrict__ out, int G) {
  int gi = blockIdx.x * blockDim.x + threadIdx.x;
  if (gi >= G) return;
  float s = lb3[0];
#pragma unroll
  for (int c = 0; c < HIDC; ++c) s += g2[(size_t)gi * HIDC + c] * lw3[c];
  out[gi] = s > 0.0f ? s : 0.0f;
}

// ---------------------------------------------------------------------------
extern "C" void kernel_launch(void* const* d_in, const int* in_sizes, int n_in,
                              void* d_out, int out_size, void* d_ws, size_t ws_size,
                              hipStream_t stream) {
  const float* x     = (const float*)d_in[0];
  const int*   ei    = (const int*)d_in[1];
  const float* ef    = (const float*)d_in[2];
  const int*   batch = (const int*)d_in[3];

  const int N  = in_sizes[0] / NODEF;      // 50000
  const int E  = in_sizes[1] / 2;          // 800000
  const int E2 = E + N;                    // with self-loops
  const int G  = out_size;                 // 2048

  // workspace carve-up (floats)
  float* wsf = (float*)d_ws;
  size_t off = 0;
  float*    bufH  = wsf + off; off += (size_t)N * FEAT;
  float*    bufO  = wsf + off; off += (size_t)N * FEAT;
  float*    bufA  = wsf + off; off += (size_t)N * FEAT;
  float*    xp    = wsf + off; off += (size_t)N * NODEFP;
  float*    w1p   = wsf + off; off += (size_t)NODEFP * FEAT;
  float*    als   = wsf + off; off += (size_t)N * HEADS;
  float*    ald   = wsf + off; off += (size_t)N * HEADS;
  float*    ale   = wsf + off; off += (size_t)E * HEADS;
  float*    ex    = wsf + off; off += (size_t)E2 * HEADS;
  unsigned* mx    = (unsigned*)(wsf + off); off += (size_t)N * HEADS;
  float*    den   = wsf + off; off += (size_t)N * HEADS;
  float*    meanE = wsf + off; off += 8;
  float*    Wea   = wsf + off; off += 32;
  float*    selfv = wsf + off; off += 4;
  float*    g     = wsf + off; off += (size_t)G * FEAT;
  float*    g1    = wsf + off; off += (size_t)G * HIDC;
  float*    g2    = wsf + off; off += (size_t)G * HIDC;
  (void)ws_size; (void)n_in;

  auto cdiv = [](long long a, long long b) { return (unsigned)((a + b - 1) / b); };

  gat_mean_edge<<<8, 256, 0, stream>>>(ef, meanE, E);
  gat_pad_x<<<cdiv((long long)N * NODEFP, 256), 256, 0, stream>>>(x, xp, N);
  gat_pad_w1<<<cdiv(NODEFP * FEAT, 256), 256, 0, stream>>>((const float*)d_in[4], w1p);

  for (int l = 0; l < 3; ++l) {
    int base = 4 + l * 6;
    const float* W    = (l == 0) ? w1p : (const float*)d_in[base + 0];
    const float* asrc = (const float*)d_in[base + 1];
    const float* adst = (const float*)d_in[base + 2];
    const float* We   = (const float*)d_in[base + 3];
    const float* ae   = (const float*)d_in[base + 4];
    const float* b    = (const float*)d_in[base + 5];
    const float* in   = (l == 0) ? xp : ((l == 1) ? bufO : bufA);
    float*       agg  = (l == 1) ? bufA : bufO;
    const int    K    = (l == 0) ? NODEFP : FEAT;

    gat_layer_prep<<<1, 32, 0, stream>>>(We, ae, meanE, Wea, selfv);

    // node GEMM: (50000/16)=3125 row tiles, panel factor 5 divides exactly
    long long tiles_node = ((long long)(N / 16) / MT_NODE) * (FEAT / 16);
    gat_wmma_gemm_f32<MT_NODE><<<cdiv(tiles_node, 8), 256, 0, stream>>>(
        in, W, bufH, nullptr, N, K, FEAT, 0);

    gat_node_dots<<<cdiv(N, 256), 256, 0, stream>>>(bufH, asrc, adst, als, ald, N);
    gat_edge_ale<<<cdiv(E, 256), 256, 0, stream>>>(ef, Wea, ale, E);
    gat_init_layer<<<cdiv((long long)N * FEAT, 256), 256, 0, stream>>>(agg, mx, den, N);
    gat_edge_max<<<cdiv(E2, 256), 256, 0, stream>>>(ei, als, ald, ale, selfv, ex, mx, E, E2);
    gat_edge_exp<<<cdiv(E2, 256), 256, 0, stream>>>(ei, ex, mx, den, E, E2);
    gat_edge_msg<<<cdiv((long long)E2 * FEAT, 256), 256, 0, stream>>>(ei, bufH, ex, den, agg, E, E2);
    gat_bias_relu<<<cdiv((long long)N * FEAT, 256), 256, 0, stream>>>(agg, b, N);
  }

  // pooling: final layer aggregate lives in bufO (layers 0,2 -> bufO)
  gat_zero<<<cdiv((long long)G * FEAT, 256), 256, 0, stream>>>(g, G * FEAT);
  gat_pool<<<cdiv((long long)N * FEAT, 256), 256, 0, stream>>>(bufO, batch, g, N);

  const float* lw1 = (const float*)d_in[22];
  const float* lb1 = (const float*)d_in[23];
  const float* lw2 = (const float*)d_in[24];
  const float* lb2 = (const float*)d_in[25];
  const float* lw3 = (const float*)d_in[26];
  const float* lb3 = (const float*)d_in[27];

  // MLP GEMMs: (2048/16)=128 row tiles, panel factor 4 divides exactly
  long long tiles_mlp = ((long long)(G / 16) / MT_MLP) * (HIDC / 16);
  gat_wmma_gemm_f32<MT_MLP><<<cdiv(tiles_mlp, 8), 256, 0, stream>>>(
      g,  lw1, g1, lb1, G, FEAT, HIDC, 1);
  gat_wmma_gemm_f32<MT_MLP><<<cdiv(tiles_mlp, 8), 256, 0, stream>>>(
      g1, lw2, g2, lb2, G, HIDC, HIDC, 1);
  gat_mlp_out<<<cdiv(G, 256), 256, 0, stream>>>(g2, lw3, lb3, (float*)d_out, G);
}